// GSubgroupKernel_65884798320719
// MI455X (gfx1250) — compile-verified
//
#include <hip/hip_runtime.h>

typedef __attribute__((ext_vector_type(2))) float v2f;
typedef __attribute__((ext_vector_type(4))) float v4f;
typedef __attribute__((ext_vector_type(8))) float v8f;

#define OUT_C  384
#define IN_C   384
#define KGRID  16
#define NUM_IN 16
#define NUM_OUT 16

static constexpr float TWO_PI_F = 6.28318530717958647692f;

// out[oc][ni][ic][no] = (1-frac)*weight[oc][ic][i0] + frac*weight[oc][ic][i1]
// with (i0,i1,frac) a function of (ni,no) only: circular lerp of angle
// mod(in_H[ni]-out_H[no], 2pi) on the uniform K=16 grid.
//
// D = C x W per 16x16 tile via V_WMMA_F32_16X16X4_F32 (x4 for K=16):
//   A-slot (16x4): C coefficients, lane = M = no,  VGPR0/1 = K pair (hi half: K+2)
//   B-slot (4x16): weight tile,    lane = N = oc,  VGPR0/1 = K pair (contiguous in memory)
//   D (16x16):     lane = N = oc,  VGPR r = M = no (+8 for lanes 16..31)
//
// Each wave owns TWO ic columns so the (ni,no,k) coefficient fragments
// (pure VALU) are reused across both WMMA streams. Output stores are
// non-temporal (streaming, write-once).
__global__ __launch_bounds__(256) void gsub_wmma_kernel(
    const float* __restrict__ in_H,
    const float* __restrict__ out_H,
    const float* __restrict__ weight,
    float* __restrict__ out)
{
    const int lane  = threadIdx.x & 31;
    const int wave  = threadIdx.x >> 5;
    const int laneN = lane & 15;          // A-row (no) AND B/D-column (oc_local)
    const bool hi   = lane >= 16;

    const int oc_base = blockIdx.x * 16;                 // 24 tiles over OUT_C
    const int ic0     = (blockIdx.y * 8 + wave) * 2;     // 24 blocks x 8 waves x 2 ic = 384
    const int ic1     = ic0 + 1;

    // ---- B-slot fragments: weight[(oc_base+laneN)][ic][k], K-pairs per VGPR ----
    const float* wrow = weight
        + (size_t)(oc_base + laneN) * (IN_C * KGRID)
        + (hi ? 2 : 0);
    const float* wp0 = wrow + (size_t)ic0 * KGRID;
    const float* wp1 = wrow + (size_t)ic1 * KGRID;
    v2f wa0 = *(const v2f*)(wp0 + 0);   // ic0: K = {0,1} / {2,3}
    v2f wa1 = *(const v2f*)(wp0 + 4);   //      K = {4,5} / {6,7}
    v2f wa2 = *(const v2f*)(wp0 + 8);
    v2f wa3 = *(const v2f*)(wp0 + 12);
    v2f wb0 = *(const v2f*)(wp1 + 0);   // ic1
    v2f wb1 = *(const v2f*)(wp1 + 4);
    v2f wb2 = *(const v2f*)(wp1 + 8);
    v2f wb3 = *(const v2f*)(wp1 + 12);

    const float outh = out_H[laneN];                     // this lane's "no" angle
    const float inv_spacing = (float)KGRID / TWO_PI_F;

    // D store bases: out[(oc_base+laneN)*NUM_IN*IN_C*NUM_OUT + ni*IN_C*NUM_OUT
    //                    + ic*NUM_OUT + (no = r + (hi?8:0))]
    float* orow = out
        + (size_t)(oc_base + laneN) * (NUM_IN * IN_C * NUM_OUT)
        + (hi ? 8 : 0);
    float* ob0 = orow + (size_t)ic0 * NUM_OUT;
    float* ob1 = orow + (size_t)ic1 * NUM_OUT;

    for (int ni = 0; ni < NUM_IN; ++ni) {
        // circular-lerp coefficients for (ni, no=laneN)
        float t    = (in_H[ni] - outh) * inv_spacing;    // in (-16, 16)
        float fl   = floorf(t);
        float frac = t - fl;                             // invariant under +16
        int   i0   = (((int)fl) + 16) & 15;              // == mod(floor, 16)
        int   i1   = (i0 + 1) & 15;
        float w0   = 1.0f - frac;
        float w1   = frac;

        // A-slot coefficient fragments, one per K-slice (shared by both ic)
        v2f cf[4];
        #pragma unroll
        for (int kb = 0; kb < 4; ++kb) {
            const int k0 = kb * 4 + (hi ? 2 : 0);
            const int k1 = k0 + 1;
            cf[kb].x = (k0 == i0) ? w0 : ((k0 == i1) ? w1 : 0.0f);
            cf[kb].y = (k1 == i0) ? w0 : ((k1 == i1) ? w1 : 0.0f);
        }

        v8f c0 = {};
        v8f c1 = {};
        c0 = __builtin_amdgcn_wmma_f32_16x16x4_f32(false, cf[0], false, wa0, (short)0, c0, false, false);
        c1 = __builtin_amdgcn_wmma_f32_16x16x4_f32(false, cf[0], false, wb0, (short)0, c1, false, false);
        c0 = __builtin_amdgcn_wmma_f32_16x16x4_f32(false, cf[1], false, wa1, (short)0, c0, false, false);
        c1 = __builtin_amdgcn_wmma_f32_16x16x4_f32(false, cf[1], false, wb1, (short)0, c1, false, false);
        c0 = __builtin_amdgcn_wmma_f32_16x16x4_f32(false, cf[2], false, wa2, (short)0, c0, false, false);
        c1 = __builtin_amdgcn_wmma_f32_16x16x4_f32(false, cf[2], false, wb2, (short)0, c1, false, false);
        c0 = __builtin_amdgcn_wmma_f32_16x16x4_f32(false, cf[3], false, wa3, (short)0, c0, false, false);
        c1 = __builtin_amdgcn_wmma_f32_16x16x4_f32(false, cf[3], false, wb3, (short)0, c1, false, false);

        // 8 consecutive floats per lane per ic -> non-temporal b128 stores
        float* o0 = ob0 + (size_t)ni * (IN_C * NUM_OUT);
        float* o1 = ob1 + (size_t)ni * (IN_C * NUM_OUT);
        v4f c0lo = { c0[0], c0[1], c0[2], c0[3] };
        v4f c0hi = { c0[4], c0[5], c0[6], c0[7] };
        v4f c1lo = { c1[0], c1[1], c1[2], c1[3] };
        v4f c1hi = { c1[4], c1[5], c1[6], c1[7] };
        __builtin_nontemporal_store(c0lo, (v4f*)(o0 + 0));
        __builtin_nontemporal_store(c0hi, (v4f*)(o0 + 4));
        __builtin_nontemporal_store(c1lo, (v4f*)(o1 + 0));
        __builtin_nontemporal_store(c1hi, (v4f*)(o1 + 4));
    }
}

extern "C" void kernel_launch(void* const* d_in, const int* in_sizes, int n_in,
                              void* d_out, int out_size, void* d_ws, size_t ws_size,
                              hipStream_t stream) {
    const float* in_H   = (const float*)d_in[0];
    const float* out_H  = (const float*)d_in[1];
    const float* weight = (const float*)d_in[2];
    // d_in[3] = grid_H: uniform grid, spacing known statically -> unused
    float* out = (float*)d_out;

    dim3 grid(OUT_C / 16, IN_C / 16, 1);   // (24, 24); 8 waves x 2 ic per block
    gsub_wmma_kernel<<<grid, 256, 0, stream>>>(in_H, out_H, weight, out);
}